// GCN_72567767433973
// MI455X (gfx1250) — compile-verified
//
#include <hip/hip_runtime.h>

#define N_NODES 50000
#define N_EDGES 800000
#define N_GRAPHS 50
#define D 128
#define DOUT 10
#define MTILES (N_NODES / 16)   // 3125 exactly

typedef __attribute__((ext_vector_type(2))) float v2f;
typedef __attribute__((ext_vector_type(8))) float v8f;

// ---------------- init ----------------
__global__ void gcn_zero(float* __restrict__ p, int n) {
    int i = blockIdx.x * blockDim.x + threadIdx.x;
    if (i < n) p[i] = 0.0f;
}

// ---------------- degrees -> norm ----------------
__global__ void gcn_degrees(const int* __restrict__ src, const int* __restrict__ dst,
                            float* __restrict__ degs, float* __restrict__ degd) {
    int i = blockIdx.x * blockDim.x + threadIdx.x;
    if (i < N_EDGES) {
        atomicAdd(&degs[src[i]], 1.0f);
        atomicAdd(&degd[dst[i]], 1.0f);
    }
}

__global__ void gcn_to_norm(float* __restrict__ degs, float* __restrict__ degd) {
    int i = blockIdx.x * blockDim.x + threadIdx.x;
    if (i < N_NODES) {
        degs[i] = rsqrtf(fmaxf(degs[i], 1.0f));
        degd[i] = rsqrtf(fmaxf(degd[i], 1.0f));
    }
}

// ---------------- hs = h * norm_src ; agg = 0 ----------------
__global__ void gcn_scale_zero(const float* __restrict__ hin, const float* __restrict__ nsrc,
                               float* __restrict__ hs, float* __restrict__ agg) {
    int i = blockIdx.x * blockDim.x + threadIdx.x;   // over N_NODES*D
    if (i < N_NODES * D) {
        hs[i] = hin[i] * nsrc[i >> 7];
        agg[i] = 0.0f;
    }
}

// ---------------- edge scatter: one wave per edge, float4 per lane ----------------
__global__ void gcn_scatter(const int* __restrict__ src, const int* __restrict__ dst,
                            const float* __restrict__ hs, float* __restrict__ agg) {
    const int e = blockIdx.x * (blockDim.x >> 5) + (threadIdx.x >> 5);
    if (e >= N_EDGES) return;                 // wave-uniform
    const int lane = threadIdx.x & 31;
    const int s = src[e];
    const int d = dst[e];
    const float4 v = *reinterpret_cast<const float4*>(hs + (size_t)s * D + lane * 4);
    float* a = agg + (size_t)d * D + lane * 4;
    atomicAdd(a + 0, v.x);
    atomicAdd(a + 1, v.y);
    atomicAdd(a + 2, v.z);
    atomicAdd(a + 3, v.w);
}

// ---------------- out = relu((agg * ndst) @ W + b), fp32 WMMA 16x16x4 ----------------
// One wave per 16x16 output tile; grid = (ceil(MTILES/8), 8 n-tiles), block = 256 (8 waves).
__global__ void gcn_gemm_relu(const float* __restrict__ A, const float* __restrict__ ndst,
                              const float* __restrict__ W, const float* __restrict__ bias,
                              float* __restrict__ out) {
    const int wave  = threadIdx.x >> 5;
    const int mtile = blockIdx.x * (blockDim.x >> 5) + wave;
    if (mtile >= MTILES) return;              // wave-uniform guard (EXEC stays all-ones)
    const int ntile = blockIdx.y;
    const int lane  = threadIdx.x & 31;
    const int half  = lane >> 4;              // 0: K pair {0,1}, 1: K pair {2,3}
    const int l16   = lane & 15;

    const int arow = mtile * 16 + l16;        // A row held by this lane (both halves)
    const int col  = ntile * 16 + l16;        // B / output column held by this lane
    const float scale = ndst[arow];
    const float* Arow = A + (size_t)arow * D;

    v8f c = {};
#pragma unroll
    for (int k = 0; k < D; k += 4) {
        const float2 av = *reinterpret_cast<const float2*>(Arow + k + 2 * half);
        v2f a, b;
        a.x = av.x * scale;
        a.y = av.y * scale;
        b.x = W[(k + 2 * half)     * D + col];
        b.y = W[(k + 2 * half + 1) * D + col];
        c = __builtin_amdgcn_wmma_f32_16x16x4_f32(
                /*neg_a=*/false, a, /*neg_b=*/false, b,
                /*c_mod=*/(short)0, c, /*reuse_a=*/false, /*reuse_b=*/false);
    }

    const float bv = bias[col];
    const int rbase = mtile * 16 + half * 8;  // C layout: VGPR r -> row r (+8 for lanes 16-31)
#pragma unroll
    for (int r = 0; r < 8; ++r) {
        float v = c[r] + bv;
        v = v > 0.0f ? v : 0.0f;
        out[(size_t)(rbase + r) * D + col] = v;
    }
}

// ---------------- per-graph sum + count ----------------
__global__ void gcn_readout(const float* __restrict__ h, const int* __restrict__ gid,
                            float* __restrict__ gsum, float* __restrict__ gcnt) {
    int i = blockIdx.x * blockDim.x + threadIdx.x;   // over N_NODES*D
    if (i < N_NODES * D) {
        const int node = i >> 7;
        const int ch   = i & 127;
        const int g    = gid[node];
        atomicAdd(&gsum[g * D + ch], h[i]);
        if (ch == 0) atomicAdd(&gcnt[g], 1.0f);
    }
}

// ---------------- head: mean, small GEMM, log_softmax over axis 0 (graphs) ----------------
__global__ void gcn_head(const float* __restrict__ gsum, const float* __restrict__ gcnt,
                         const float* __restrict__ Wm, const float* __restrict__ bm,
                         float* __restrict__ out) {
    __shared__ float hg[N_GRAPHS * D];
    __shared__ float lg[N_GRAPHS * DOUT];
    __shared__ float cmax[DOUT];
    __shared__ float clse[DOUT];
    const int tid = threadIdx.x;

    for (int i = tid; i < N_GRAPHS * D; i += blockDim.x)
        hg[i] = gsum[i] / fmaxf(gcnt[i / D], 1.0f);
    __syncthreads();

    for (int i = tid; i < N_GRAPHS * DOUT; i += blockDim.x) {
        const int g = i / DOUT, o = i % DOUT;
        float acc = bm[o];
        for (int k = 0; k < D; ++k) acc += hg[g * D + k] * Wm[k * DOUT + o];
        lg[i] = acc;
    }
    __syncthreads();

    if (tid < DOUT) {
        float m = -3.0e38f;
        for (int g = 0; g < N_GRAPHS; ++g) m = fmaxf(m, lg[g * DOUT + tid]);
        float s = 0.0f;
        for (int g = 0; g < N_GRAPHS; ++g) s += __expf(lg[g * DOUT + tid] - m);
        cmax[tid] = m;
        clse[tid] = logf(s);
    }
    __syncthreads();

    for (int i = tid; i < N_GRAPHS * DOUT; i += blockDim.x) {
        const int o = i % DOUT;
        out[i] = lg[i] - cmax[o] - clse[o];
    }
}

extern "C" void kernel_launch(void* const* d_in, const int* in_sizes, int n_in,
                              void* d_out, int out_size, void* d_ws, size_t ws_size,
                              hipStream_t stream) {
    const float* h   = (const float*)d_in[0];
    const int*   src = (const int*)d_in[1];
    const int*   dst = (const int*)d_in[2];
    const int*   gid = (const int*)d_in[3];
    const float* W1  = (const float*)d_in[4];
    const float* b1  = (const float*)d_in[5];
    const float* W2  = (const float*)d_in[6];
    const float* b2  = (const float*)d_in[7];
    const float* W3  = (const float*)d_in[8];
    const float* b3  = (const float*)d_in[9];
    const float* Wm  = (const float*)d_in[10];
    const float* bm  = (const float*)d_in[11];
    float* out = (float*)d_out;

    // workspace layout (floats)
    float* ws   = (float*)d_ws;
    float* nsrc = ws;                              // 50048 (padded)
    float* ndst = nsrc + 50048;                    // 50048
    float* hs   = ndst + 50048;                    // N*D
    float* agg  = hs   + (size_t)N_NODES * D;      // N*D
    float* hb   = agg  + (size_t)N_NODES * D;      // N*D
    float* gsum = hb   + (size_t)N_NODES * D;      // 6400
    float* gcnt = gsum + N_GRAPHS * D;             // 50 (contiguous after gsum)

    const int B = 256;
    const int ND = N_NODES * D;                    // 6,400,000

    // zero the atomically-accumulated buffers
    gcn_zero<<<(2 * 50048 + B - 1) / B, B, 0, stream>>>(nsrc, 2 * 50048);
    gcn_zero<<<(N_GRAPHS * D + N_GRAPHS + B - 1) / B, B, 0, stream>>>(gsum, N_GRAPHS * D + N_GRAPHS);

    // degree -> norm
    gcn_degrees<<<(N_EDGES + B - 1) / B, B, 0, stream>>>(src, dst, nsrc, ndst);
    gcn_to_norm<<<(N_NODES + B - 1) / B, B, 0, stream>>>(nsrc, ndst);

    const dim3 gemm_grid((MTILES + 7) / 8, 8, 1);
    const int scat_blocks = N_EDGES / 8;           // 8 waves (edges) per 256-thread block

    // layer 1
    gcn_scale_zero<<<(ND + B - 1) / B, B, 0, stream>>>(h, nsrc, hs, agg);
    gcn_scatter<<<scat_blocks, B, 0, stream>>>(src, dst, hs, agg);
    gcn_gemm_relu<<<gemm_grid, B, 0, stream>>>(agg, ndst, W1, b1, hb);

    // layer 2 (hb consumed into hs before being overwritten; stream-ordered)
    gcn_scale_zero<<<(ND + B - 1) / B, B, 0, stream>>>(hb, nsrc, hs, agg);
    gcn_scatter<<<scat_blocks, B, 0, stream>>>(src, dst, hs, agg);
    gcn_gemm_relu<<<gemm_grid, B, 0, stream>>>(agg, ndst, W2, b2, hb);

    // layer 3
    gcn_scale_zero<<<(ND + B - 1) / B, B, 0, stream>>>(hb, nsrc, hs, agg);
    gcn_scatter<<<scat_blocks, B, 0, stream>>>(src, dst, hs, agg);
    gcn_gemm_relu<<<gemm_grid, B, 0, stream>>>(agg, ndst, W3, b3, hb);

    // readout + head
    gcn_readout<<<(ND + B - 1) / B, B, 0, stream>>>(hb, gid, gsum, gcnt);
    gcn_head<<<1, 512, 0, stream>>>(gsum, gcnt, Wm, bm, out);
}